// BahdanauAttention_softmax_71768903517076
// MI455X (gfx1250) — compile-verified
//
#include <hip/hip_runtime.h>
#include <math.h>

typedef __attribute__((ext_vector_type(2))) float v2f;
typedef __attribute__((ext_vector_type(8))) float v8f;

#define B_  64
#define N1_ 8
#define K_  128
#define E_  256
#define H_  512
#define LDK 260   // padded LDS row stride in floats: 260 % 64 == 4 -> conflict-free b64 row fetches

// Branchless tanh: tanh(x) = 1 - 2/(exp(2x)+1).
// exp2 hw op saturates to +inf / 0 for large |x| -> result saturates to +/-1 with no branches.
__device__ __forceinline__ float fast_tanhf(float x) {
    float e = __builtin_amdgcn_exp2f(x * 2.8853900817779268f);  // 2*log2(e)
    float r = __builtin_amdgcn_rcpf(e + 1.0f);
    return fmaf(-2.0f, r, 1.0f);
}

__global__ __launch_bounds__(256)
void bahdanau_attn_kernel(const float* __restrict__ queries,
                          const float* __restrict__ keys,
                          const int*   __restrict__ masks,
                          const float* __restrict__ wk_w,
                          const float* __restrict__ wk_b,
                          const float* __restrict__ wq_w,
                          const float* __restrict__ wq_b,
                          const float* __restrict__ wv_w,
                          const float* __restrict__ wv_b,
                          float* __restrict__ out_attn,   // (B,N1,E)
                          float* __restrict__ out_wt)     // (B,N1,K,1)
{
    __shared__ float keysLds[K_ * LDK];      // 130 KB
    __shared__ float panelLds[16 * LDK];     // 16.25 KB
    __shared__ float qkLds[H_];              // q_proj + wq_b + wk_b
    __shared__ float wvLds[H_];
    __shared__ float scoreLds[K_];
    __shared__ float red[K_];
    __shared__ float sq[E_];

    const int tid  = threadIdx.x;
    const int lane = tid & 31;
    const int wave = tid >> 5;
    const int bn   = blockIdx.x;                       // b*N1 + n
    const size_t keyBase = (size_t)bn * K_ * E_;

    // ---------------- stage q and keys block into LDS ----------------
    if (tid < E_) sq[tid] = queries[(size_t)bn * E_ + tid];
    {
        const float4* ksrc = (const float4*)(keys + keyBase);
        for (int i = tid; i < (K_ * E_) / 4; i += 256) {
            int kk = i >> 6;                // 64 float4 per 256-float row
            int e4 = i & 63;
            float4 v = ksrc[i];
            *(float4*)&keysLds[kk * LDK + e4 * 4] = v;
        }
    }
    __syncthreads();

    // ---------------- q projection: qk[h] = q.wq_w[h] + wq_b[h] + wk_b[h] ----------------
    for (int h = tid; h < H_; h += 256) {
        float s = wq_b[h] + wk_b[h];
        const float* wrow = wq_w + (size_t)h * E_;
        #pragma unroll 8
        for (int e = 0; e < E_; ++e) s = fmaf(sq[e], wrow[e], s);
        qkLds[h] = s;
        wvLds[h] = wv_w[h];
    }

    // ---------------- fused keys @ wk_w^T GEMM + tanh + score reduce ----------------
    const int m0   = wave * 16;        // this wave's 16 key rows
    const int half = lane >> 4;
    const int l16  = lane & 15;
    const int arow = (m0 + l16) * LDK + 2 * half;   // A: keys rows, K-cols {e0+2h, e0+2h+1}
    const int brow = l16 * LDK + 2 * half;          // B: panel rows (h), same K split

    float scoreAcc[8];
    #pragma unroll
    for (int r = 0; r < 8; ++r) scoreAcc[r] = 0.f;

    for (int n0 = 0; n0 < H_; n0 += 16) {
        __syncthreads();   // prior panel fully consumed (and qkLds ready on iter 0)
        // cooperative load of wk_w rows n0..n0+15 into LDS panel
        {
            const float4* wsrc = (const float4*)(wk_w + (size_t)n0 * E_);
            for (int i = tid; i < (16 * E_) / 4; i += 256) {
                int hh = i >> 6;
                int e4 = i & 63;
                *(float4*)&panelLds[hh * LDK + e4 * 4] = wsrc[i];
            }
        }
        __syncthreads();

        v8f acc = {0.f, 0.f, 0.f, 0.f, 0.f, 0.f, 0.f, 0.f};
        #pragma unroll 16
        for (int e0 = 0; e0 < E_; e0 += 4) {
            v2f a = *(const v2f*)&keysLds[arow + e0];
            v2f b = *(const v2f*)&panelLds[brow + e0];
            acc = __builtin_amdgcn_wmma_f32_16x16x4_f32(
                false, a, false, b, (short)0, acc, false, false);
        }

        // epilogue in C-layout: VGPR r, lanes 0-15 -> M=r, lanes 16-31 -> M=8+r; N = n0 + l16.
        // Accumulate per-lane partials only; cross-lane reduction is hoisted out of the loop
        // (lanes of a half-wave cover disjoint h across all panels).
        const int   h  = n0 + l16;
        const float qk = qkLds[h];
        const float wv = wvLds[h];
        #pragma unroll
        for (int r = 0; r < 8; ++r) {
            float t = fast_tanhf(acc[r] + qk);
            scoreAcc[r] = fmaf(t, wv, scoreAcc[r]);
        }
    }

    // single width-16 butterfly reduction per row, after all panels
    #pragma unroll
    for (int r = 0; r < 8; ++r) {
        float p = scoreAcc[r];
        p += __shfl_xor(p, 1, 16);
        p += __shfl_xor(p, 2, 16);
        p += __shfl_xor(p, 4, 16);
        p += __shfl_xor(p, 8, 16);
        scoreAcc[r] = p;
    }

    const float wvb = wv_b[0];
    if (l16 == 0) {
        const int rowbase = m0 + (half ? 8 : 0);
        #pragma unroll
        for (int r = 0; r < 8; ++r)
            scoreLds[rowbase + r] = scoreAcc[r] + wvb;
    }
    __syncthreads();

    // ---------------- masked softmax over K ----------------
    float sv = -INFINITY, ev = 0.f;
    if (tid < K_) {
        sv = scoreLds[tid] + (float)masks[(size_t)bn * K_ + tid] * -1e9f;
        red[tid] = sv;
    }
    __syncthreads();
    for (int s = 64; s > 0; s >>= 1) {
        if (tid < s) red[tid] = fmaxf(red[tid], red[tid + s]);
        __syncthreads();
    }
    const float mx = red[0];
    __syncthreads();
    if (tid < K_) { ev = expf(sv - mx); red[tid] = ev; }
    __syncthreads();
    for (int s = 64; s > 0; s >>= 1) {
        if (tid < s) red[tid] += red[tid + s];
        __syncthreads();
    }
    const float denom = red[0];
    __syncthreads();
    if (tid < K_) {
        float w = ev / denom;
        scoreLds[tid] = w;
        out_wt[(size_t)bn * K_ + tid] = w;
    }
    __syncthreads();

    // ---------------- attn_out[e] = sum_k w[k] * keys[k][e] ----------------
    {
        float o = 0.f;
        #pragma unroll 8
        for (int k = 0; k < K_; ++k)
            o = fmaf(scoreLds[k], keysLds[k * LDK + tid], o);
        out_attn[(size_t)bn * E_ + tid] = o;
    }
}

extern "C" void kernel_launch(void* const* d_in, const int* in_sizes, int n_in,
                              void* d_out, int out_size, void* d_ws, size_t ws_size,
                              hipStream_t stream) {
    const float* queries = (const float*)d_in[0];
    const float* keys    = (const float*)d_in[1];
    const int*   masks   = (const int*)d_in[2];
    // d_in[3] = num_neg (unused by the reference math)
    const float* wk_w = (const float*)d_in[4];
    const float* wk_b = (const float*)d_in[5];
    const float* wq_w = (const float*)d_in[6];
    const float* wq_b = (const float*)d_in[7];
    const float* wv_w = (const float*)d_in[8];
    const float* wv_b = (const float*)d_in[9];

    float* out_attn = (float*)d_out;                         // (B,N1,E) first
    float* out_wt   = out_attn + (size_t)B_ * N1_ * E_;      // then (B,N1,K,1)

    dim3 grid(B_ * N1_);   // 512 workgroups, one per (b, n)
    bahdanau_attn_kernel<<<grid, 256, 0, stream>>>(
        queries, keys, masks, wk_w, wk_b, wq_w, wq_b, wv_w, wv_b,
        out_attn, out_wt);
}